// CaDistogramLoss_45002667327960
// MI455X (gfx1250) — compile-verified
//
#include <hip/hip_runtime.h>
#include <math.h>

typedef __attribute__((ext_vector_type(2))) float v2f;
typedef __attribute__((ext_vector_type(8))) float v8f;

#define BATCH 2
#define NROW  512
#define DDIM  1024
#define NBINS 64

// ---------------------------------------------------------------------------
// Kernel 0: Wsum[k,d] = W[k, d] + W[k, 1024 + d]   (exploits logits symmetry)
// ---------------------------------------------------------------------------
__global__ void k_wsum(const float* __restrict__ W, float* __restrict__ wsum) {
    int idx = blockIdx.x * blockDim.x + threadIdx.x;      // 64*1024 elements
    if (idx < NBINS * DDIM) {
        int k = idx >> 10;
        int d = idx & 1023;
        wsum[idx] = W[k * (2 * DDIM) + d] + W[k * (2 * DDIM) + DDIM + d];
    }
}

// ---------------------------------------------------------------------------
// Kernel 1: s[m, k] = x[m, :] . Wsum[k, :] + bias[k]   via V_WMMA_F32_16X16X4_F32
//   M = B*N = 1024 rows, 64 cols.  One wave per 16x16 tile.
//   Block = 128 threads (4 waves), wave w owns column tile w.  Grid = 64 blocks.
// ---------------------------------------------------------------------------
__global__ void k_gemm(const float* __restrict__ x, const float* __restrict__ wsum,
                       const float* __restrict__ bias, float* __restrict__ s) {
    const int lane    = threadIdx.x & 31;
    const int wave    = threadIdx.x >> 5;
    const int rowBase = blockIdx.x * 16;
    const int colBase = wave * 16;
    const int half    = lane >> 4;     // 0: K pair {0,1}, 1: K pair {2,3}
    const int l16     = lane & 15;

    // A fragment: 16x4 f32, lane l16 = row M, VGPR0 = K0/K2, VGPR1 = K1/K3
    const float* arow = x    + (size_t)(rowBase + l16) * DDIM + half * 2;
    // B fragment: 4x16 f32 (K x N), N = l16, mirrored K striping
    const float* brow = wsum + (size_t)(colBase + l16) * DDIM + half * 2;

    v8f acc = {};
    #pragma unroll 4
    for (int kk = 0; kk < DDIM; kk += 4) {
        v2f a = *(const v2f*)(arow + kk);
        v2f b = *(const v2f*)(brow + kk);
        acc = __builtin_amdgcn_wmma_f32_16x16x4_f32(
            /*neg_a=*/false, a, /*neg_b=*/false, b,
            /*c_mod=*/(short)0, acc, /*reuse_a=*/false, /*reuse_b=*/false);
    }

    // D layout: lanes 0-15 -> M = v, lanes 16-31 -> M = 8+v; N = l16
    const int   col = colBase + l16;
    const float bv  = bias[col];
    #pragma unroll
    for (int v = 0; v < 8; ++v) {
        int row = rowBase + v + half * 8;
        s[row * NBINS + col] = acc[v] + bv;
    }
}

// ---------------------------------------------------------------------------
// Kernel 2: per (b,i) row of the pair matrix, fused softmax-CE + mask + reduce.
//   Grid = B*N blocks of 256 threads; thread t handles j = t and j = t+256.
// ---------------------------------------------------------------------------
__global__ void k_pairs(const float* __restrict__ s, const float* __restrict__ A,
                        const unsigned char* __restrict__ pad,
                        float* __restrict__ rowsum) {
    __shared__ float si[NBINS];
    __shared__ float cai[3];
    __shared__ float red[256];

    const int bi  = blockIdx.x;          // b*512 + i
    const int b   = bi >> 9;
    const int i   = bi & (NROW - 1);
    const int tid = threadIdx.x;

    if (pad[bi]) {                        // whole row masked out
        if (tid == 0) rowsum[bi] = 0.0f;
        return;
    }

    const float* caBase = A + (size_t)b * (3 * NROW * 3) + (NROW * 3); // A[b,1,:,:]
    if (tid < NBINS) si[tid]  = s[(size_t)bi * NBINS + tid];
    if (tid < 3)     cai[tid] = caBase[i * 3 + tid];
    __syncthreads();

    float acc = 0.0f;
    #pragma unroll
    for (int jj = 0; jj < NROW; jj += 256) {
        const int j  = jj + tid;
        const int bj = b * NROW + j;
        if (!pad[bj]) {
            const float* sj = s + (size_t)bj * NBINS;

            float vreg[NBINS];
            float m = -3.402823466e+38f;
            #pragma unroll
            for (int k = 0; k < NBINS; ++k) {
                float v = si[k] + sj[k];
                vreg[k] = v;
                m = fmaxf(m, v);
            }
            float sum = 0.0f;
            #pragma unroll
            for (int k = 0; k < NBINS; ++k)
                sum += __expf(vreg[k] - m);

            // true bin: boundaries are (2.3125 + t*0.3125)^2, strict-less count
            const float* caj = caBase + j * 3;
            float dx = cai[0] - caj[0];
            float dy = cai[1] - caj[1];
            float dz = cai[2] - caj[2];
            float r  = sqrtf(dx * dx + dy * dy + dz * dz);
            int   t  = (int)ceilf((r - 2.3125f) * 3.2f);
            t = t < 0 ? 0 : (t > NBINS - 1 ? NBINS - 1 : t);

            float lse = m + __logf(sum);
            acc += lse - (si[t] + sj[t]);   // direct reads: no dynamic reg index
        }
    }

    red[tid] = acc;
    __syncthreads();
    #pragma unroll
    for (int off = 128; off > 0; off >>= 1) {
        if (tid < off) red[tid] += red[tid + off];
        __syncthreads();
    }
    if (tid == 0) rowsum[bi] = red[0];
}

// ---------------------------------------------------------------------------
// Kernel 3: final reduction.  denom_b = 1e-6 + (N - P_b)^2 ; mean over batch.
// ---------------------------------------------------------------------------
__global__ void k_final(const float* __restrict__ rowsum,
                        const unsigned char* __restrict__ pad,
                        float* __restrict__ out) {
    __shared__ float rs0[512], rs1[512];
    __shared__ int   pc0[512], pc1[512];
    const int t = threadIdx.x;            // 512 threads
    rs0[t] = rowsum[t];
    rs1[t] = rowsum[NROW + t];
    pc0[t] = pad[t] ? 1 : 0;
    pc1[t] = pad[NROW + t] ? 1 : 0;
    __syncthreads();
    #pragma unroll
    for (int off = 256; off > 0; off >>= 1) {
        if (t < off) {
            rs0[t] += rs0[t + off];  rs1[t] += rs1[t + off];
            pc0[t] += pc0[t + off];  pc1[t] += pc1[t + off];
        }
        __syncthreads();
    }
    if (t == 0) {
        float n0 = (float)NROW - (float)pc0[0];
        float n1 = (float)NROW - (float)pc1[0];
        float d0 = 1e-6f + n0 * n0;
        float d1 = 1e-6f + n1 * n1;
        out[0] = 0.5f * (rs0[0] / d0 + rs1[0] / d1);
    }
}

// ---------------------------------------------------------------------------
extern "C" void kernel_launch(void* const* d_in, const int* in_sizes, int n_in,
                              void* d_out, int out_size, void* d_ws, size_t ws_size,
                              hipStream_t stream) {
    const float*         x   = (const float*)d_in[0];         // (B,N,D)
    const float*         A   = (const float*)d_in[1];         // (B,3,N,3)
    const unsigned char* pad = (const unsigned char*)d_in[2]; // (B,N) bool
    const float*         W   = (const float*)d_in[3];         // (64, 2D)
    const float*         bia = (const float*)d_in[4];         // (64,)

    float* wsum   = (float*)d_ws;                 // 64*1024 f32 = 256 KB
    float* s      = wsum + NBINS * DDIM;          // 1024*64 f32 = 256 KB
    float* rowsum = s + BATCH * NROW * NBINS;     // 1024 f32

    k_wsum <<<(NBINS * DDIM + 255) / 256, 256, 0, stream>>>(W, wsum);
    k_gemm <<<(BATCH * NROW) / 16, 128, 0, stream>>>(x, wsum, bia, s);
    k_pairs<<<BATCH * NROW, 256, 0, stream>>>(s, A, pad, rowsum);
    k_final<<<1, 512, 0, stream>>>(rowsum, pad, (float*)d_out);
}